// FeatureConverge_82145544503995
// MI455X (gfx1250) — compile-verified
//
#include <hip/hip_runtime.h>
#include <hip/hip_bf16.h>
#include <math.h>

typedef float v2f __attribute__((ext_vector_type(2)));
typedef float v8f __attribute__((ext_vector_type(8)));
typedef unsigned int u32x4 __attribute__((ext_vector_type(4)));
typedef int i32x8 __attribute__((ext_vector_type(8)));
typedef int i32x4 __attribute__((ext_vector_type(4)));

#define B_ 64
#define S_ 2048
#define C_ 128
#define G_ 64
#define K_ 128

// Output concatenation (reference returns (x_bag, Asite_x, x_site)):
#define OUT_XBAG_OFF  0
#define OUT_ASITE_OFF (B_*G_*K_*C_)           // 67,108,864
#define OUT_XSITE_OFF (OUT_ASITE_OFF + B_*C_) // +8,192

// ---------------------------------------------------------------------------
// Weight-norm: w1[co,ci,t] = g1[co] * v1[co,ci,t] / ||v1[co,:,:]||
// ---------------------------------------------------------------------------
__global__ void wnorm_kernel(const float* __restrict__ v1,
                             const float* __restrict__ g1,
                             float* __restrict__ w1) {
    int co = blockIdx.x;
    __shared__ float red[128];
    float s = 0.f;
    for (int i = threadIdx.x; i < C_ * 3; i += 128) {
        float v = v1[co * C_ * 3 + i];
        s += v * v;
    }
    red[threadIdx.x] = s;
    __syncthreads();
    for (int off = 64; off > 0; off >>= 1) {
        if ((int)threadIdx.x < off) red[threadIdx.x] += red[threadIdx.x + off];
        __syncthreads();
    }
    float scale = g1[co] * rsqrtf(red[0]);
    for (int i = threadIdx.x; i < C_ * 3; i += 128)
        w1[co * C_ * 3 + i] = v1[co * C_ * 3 + i] * scale;
}

// ---------------------------------------------------------------------------
// conv1 (kernel-3 1D conv C->C along S) + tanh(tanh()) -> x_site (B,C,S)
// Block: 16 s-positions x 128 co.  8 waves, each owns one 16-wide co tile.
// M = s-offset (16), N = co (16), K = (ci,t) pairs, stepped 4 at a time.
// ---------------------------------------------------------------------------
__global__ __launch_bounds__(256) void conv1_kernel(
    const float* __restrict__ x, const float* __restrict__ w1,
    const float* __restrict__ b1, float* __restrict__ xsite) {
    __shared__ float xs[18 * C_];   // rows s0-1 .. s0+16, zero-padded per batch
    int b  = blockIdx.x / (S_ / 16);
    int s0 = (blockIdx.x % (S_ / 16)) * 16;

    for (int i = threadIdx.x; i < 18 * C_; i += 256) {
        int r = i >> 7, c = i & (C_ - 1);
        int sg = s0 + r - 1;
        xs[i] = (sg >= 0 && sg < S_) ? x[(b * S_ + sg) * C_ + c] : 0.f;
    }
    __syncthreads();

    int lane = threadIdx.x & 31;
    int wv   = threadIdx.x >> 5;
    int co0  = wv * 16;
    int m    = lane & 15;              // A-matrix row
    int k0   = (lane >> 4) << 1;       // K pair base: 0 (lanes 0-15), 2 (16-31)
    int n    = lane & 15;              // B-matrix column

    v8f acc = {};
    for (int t = 0; t < 3; ++t) {
        const float* xrow = &xs[(m + t) * C_];   // s = s0 + m + t - 1
        for (int ci0 = 0; ci0 < C_; ci0 += 4) {
            v2f a, bb;
            a.x = xrow[ci0 + k0];
            a.y = xrow[ci0 + k0 + 1];
            const float* wp = w1 + (co0 + n) * (C_ * 3) + (ci0 + k0) * 3 + t;
            bb.x = wp[0];
            bb.y = wp[3];
            acc = __builtin_amdgcn_wmma_f32_16x16x4_f32(
                false, a, false, bb, (short)0, acc, false, false);
        }
    }

    int mbase = (lane >> 4) << 3;      // C/D rows: v (lanes 0-15), v+8 (16-31)
    int co = co0 + n;
    float bias = b1[co];
    float r[8];
#pragma unroll
    for (int v = 0; v < 8; ++v) r[v] = tanhf(tanhf(acc[v] + bias));
    // 8 consecutive s positions per lane -> two b128 stores
    size_t base = (size_t)(b * C_ + co) * S_ + s0 + mbase;
    *(float4*)&xsite[base]     = make_float4(r[0], r[1], r[2], r[3]);
    *(float4*)&xsite[base + 4] = make_float4(r[4], r[5], r[6], r[7]);
}

// ---------------------------------------------------------------------------
// Asite_x = x[:, 20, :]
// ---------------------------------------------------------------------------
__global__ void asite_kernel(const float* __restrict__ x,
                             float* __restrict__ out) {
    int i = blockIdx.x * 256 + threadIdx.x;
    if (i < B_ * C_) {
        int b = i / C_, c = i % C_;
        out[i] = x[(b * S_ + 20) * C_ + c];
    }
}

// ---------------------------------------------------------------------------
// gather + conv2 (3x3, G->G over (K,C) plane) + bias -> x_bag (B,G,K,C)
// Block: one (b,k).  Context rows are gathered with the Tensor Data Mover in
// GATHER mode: one tensor_load_to_lds moves 16 index-addressed rows of
// 128 f32 from x[b] (an S_ x C_ tensor) straight into LDS (TENSORcnt).
// Each wave then owns one 16-wide c tile and accumulates all 4 go-tiles so
// the LDS B-fragment is reused 4x per K-step.  K = (dk,dc,gi): 144 steps.
// ---------------------------------------------------------------------------
__global__ __launch_bounds__(256) void conv2_kernel(
    const float* __restrict__ x, const int* __restrict__ index_all,
    const float* __restrict__ W2, const float* __restrict__ b2,
    float* __restrict__ xbag) {
    extern __shared__ float ctx[];     // [3][G_][C_] = 96 KB (dynamic LDS)
    int b = blockIdx.x / K_;
    int k = blockIdx.x % K_;
    int lane = threadIdx.x & 31;
    int wv   = threadIdx.x >> 5;

    unsigned lds_base = (unsigned)(uintptr_t)(void*)ctx;   // LDS byte offset
    unsigned long long gbase =
        (unsigned long long)(uintptr_t)(const void*)(x + (size_t)b * S_ * C_);

    // 12 groups of 16 rows; each group = one TDM gather (or zero-fill halo).
    for (int grp = wv; grp < 12; grp += 8) {
        int r0  = grp * 16;
        int kkL = r0 >> 6;             // context row block <-> conv tap dk
        int gi0 = r0 & 63;
        int kg  = k + kkL - 1;
        if (kg >= 0 && kg < K_) {
            // lanes 0-15 fetch the 16 row indices; lift to SGPRs and pack.
            int vidx = index_all[(b * G_ + gi0 + (lane & 15)) * K_ + kg];
            int id[8];
#pragma unroll
            for (int j = 0; j < 8; ++j) {
                int lo = __builtin_amdgcn_readlane(vidx, 2 * j);
                int hi = __builtin_amdgcn_readlane(vidx, 2 * j + 1);
                id[j] = (lo & 0xFFFF) | (hi << 16);
            }
            // D# group 0: count=1, gather_mode=1, 16-bit indices,
            //             lds_addr, 57-bit global_addr, type=2 ("image")
            u32x4 g0;
            g0.x = 0x80000001u;
            g0.y = lds_base + (unsigned)(r0 * C_ * 4);
            g0.z = (unsigned)(gbase & 0xFFFFFFFFull);
            g0.w = (unsigned)((gbase >> 32) & 0x1FFFFFFull) | (2u << 30);
            // D# group 1: data_size=4B, tensor 128 x 2048, tile 128 x 16 idx,
            //             dim0 stride = 128 elements
            i32x8 g1;
            g1[0] = 0x00020000;        // data_size = 2 (4 bytes)
            g1[1] = (C_ << 16);        // tensor_dim0 = 128
            g1[2] = (S_ << 16);        // tensor_dim1 = 2048
            g1[3] = (C_ << 16);        // tile_dim0  = 128
            g1[4] = 16;                // tile_dim1  = #valid indices
            g1[5] = C_;                // tensor_dim0_stride = 128
            g1[6] = 0;
            g1[7] = 0;
            // D# groups 2/3: 16 packed 16-bit row indices
            i32x4 g2 = { id[0], id[1], id[2], id[3] };
            i32x4 g3 = { id[4], id[5], id[6], id[7] };
            i32x8 g4 = {};             // reserved/extended descriptor words
            __builtin_amdgcn_tensor_load_to_lds(g0, g1, g2, g3, g4, 0);
        } else {
            float4 z = make_float4(0.f, 0.f, 0.f, 0.f);
#pragma unroll
            for (int r = 0; r < 16; ++r)
                *(float4*)&ctx[(r0 + r) * C_ + lane * 4] = z;
        }
    }
    __builtin_amdgcn_s_wait_tensorcnt(0);
    __syncthreads();

    int m  = lane & 15;
    int k0 = (lane >> 4) << 1;
    int n  = lane & 15;
    int c0 = wv * 16;

    v8f acc[4] = {};
    for (int dk = 0; dk < 3; ++dk) {
        for (int dc = 0; dc < 3; ++dc) {
            int  cc  = c0 + n + dc - 1;
            bool cok = (cc >= 0) && (cc < C_);
            int  ccc = cok ? cc : 0;   // branchless halo: EXEC stays all-1s
            for (int gi0 = 0; gi0 < G_; gi0 += 4) {
                v2f bb;
                float bv0 = ctx[(dk * G_ + gi0 + k0) * C_ + ccc];
                float bv1 = ctx[(dk * G_ + gi0 + k0 + 1) * C_ + ccc];
                bb.x = cok ? bv0 : 0.f;
                bb.y = cok ? bv1 : 0.f;
#pragma unroll
                for (int tt = 0; tt < 4; ++tt) {
                    int go0 = tt * 16;
                    v2f a;
                    const float* wp =
                        W2 + ((go0 + m) * G_ + gi0 + k0) * 9 + dk * 3 + dc;
                    a.x = wp[0];
                    a.y = wp[9];       // gi+1 (stride 9 floats in W2)
                    acc[tt] = __builtin_amdgcn_wmma_f32_16x16x4_f32(
                        false, a, false, bb, (short)0, acc[tt], false, false);
                }
            }
        }
    }

    int mbase = (lane >> 4) << 3;
    int c = c0 + n;
#pragma unroll
    for (int tt = 0; tt < 4; ++tt) {
#pragma unroll
        for (int v = 0; v < 8; ++v) {
            int go = tt * 16 + mbase + v;
            float val = acc[tt][v] + b2[go];
            xbag[((size_t)(b * G_ + go) * K_ + k) * C_ + c] = val;
        }
    }
}

// ---------------------------------------------------------------------------
extern "C" void kernel_launch(void* const* d_in, const int* in_sizes, int n_in,
                              void* d_out, int out_size, void* d_ws,
                              size_t ws_size, hipStream_t stream) {
    const float* x         = (const float*)d_in[0];
    const int*   index_all = (const int*)d_in[1];
    const float* v1        = (const float*)d_in[2];
    const float* g1        = (const float*)d_in[3];
    const float* b1        = (const float*)d_in[4];
    const float* W2        = (const float*)d_in[5];
    const float* b2        = (const float*)d_in[6];
    float* out = (float*)d_out;
    float* w1  = (float*)d_ws;   // C_*C_*3 floats = 192 KB scratch

    wnorm_kernel<<<C_, 128, 0, stream>>>(v1, g1, w1);
    conv1_kernel<<<B_ * (S_ / 16), 256, 0, stream>>>(x, w1, b1,
                                                     out + OUT_XSITE_OFF);
    asite_kernel<<<(B_ * C_ + 255) / 256, 256, 0, stream>>>(
        x, out + OUT_ASITE_OFF);
    conv2_kernel<<<B_ * K_, 256, 3 * G_ * C_ * sizeof(float), stream>>>(
        x, index_all, W2, b2, out + OUT_XBAG_OFF);
}